// AgreementRouter_37220186587187
// MI455X (gfx1250) — compile-verified
//
#include <hip/hip_runtime.h>

typedef __attribute__((ext_vector_type(2))) float v2f;
typedef __attribute__((ext_vector_type(4))) float v4f;
typedef __attribute__((ext_vector_type(8))) float v8f;

// D(16x16,f32) = A(16x4,f32) x B(4x16,f32) + C ; chained for K accumulation.
#define WMMA_F32X4(A, B, C) \
  __builtin_amdgcn_wmma_f32_16x16x4_f32(false, (A), false, (B), (short)0, (C), false, false)

// Problem constants
#define BB 64
#define NN 1152
#define CC 32
#define FF 16
#define CF 512           // C*F
#define TILES_PER_B 72   // N/16
#define SLOTS1 8         // n-split for pass-1 partial sums
#define SLOTS2 36        // wave slots per b in routing passes (2 tiles each)

// ---------------------------------------------------------------------------
// Pass 1: S[b,c,f] = sum_n in[b,n,c,f], via WMMA with A = all-ones.
// D[m,j] = sum_k 1 * B[k,j] = column sums; permutation of K rows is harmless.
// One wave per (b, c, slot); each wave sums 144 n's (36 chunks of K=4).
// ---------------------------------------------------------------------------
__global__ __launch_bounds__(256) void k_sum(const float* __restrict__ in,
                                             float* __restrict__ spart) {
  const int wid  = (blockIdx.x << 3) + (threadIdx.x >> 5);
  const int lane = threadIdx.x & 31;
  const int j    = lane & 15;
  const bool low = lane < 16;
  const int b    = wid >> 8;          // / (32*8)
  const int rem  = wid & 255;
  const int c    = rem >> 3;
  const int slot = rem & 7;

  const size_t base = (size_t)b * NN * CF + (size_t)c * FF + j;
  v8f acc = {};
  v2f ones; ones[0] = 1.0f; ones[1] = 1.0f;

  const int nbase = slot * 144 + (low ? 0 : 2);
  for (int i = 0; i < 36; ++i) {
    const int n0 = nbase + i * 4;
    v2f Bop;
    Bop[0] = in[base + (size_t)n0 * CF];        // row K (n0)
    Bop[1] = in[base + (size_t)(n0 + 1) * CF];  // row K+1 (n0+1)
    acc = WMMA_F32X4(ones, Bop, acc);
  }
  // every row of D equals the column sum; column j lives on lanes j and j+16
  if (low) spart[(((b * CC + c) * SLOTS1) + slot) * FF + j] = acc[0];
}

__global__ __launch_bounds__(256) void k_reduce_out0(const float* __restrict__ spart,
                                                     const float* __restrict__ bias,
                                                     float* __restrict__ out0) {
  const int idx = blockIdx.x * 256 + threadIdx.x;  // 32768
  const int b = idx >> 9;
  const int cf = idx & 511;
  const int c = cf >> 4, f = cf & 15;
  const size_t base = (size_t)b * (CC * SLOTS1 * FF) + c * (SLOTS1 * FF) + f;
  float s = 0.0f;
  for (int k = 0; k < SLOTS1; ++k) s += spart[base + k * FF];
  out0[idx] = s * (1.0f / 32.0f) + bias[cf];
}

// ---------------------------------------------------------------------------
// Routing pass. One wave handles (b, slot) = 2 tiles of 16 n's.
// Phase 1 (WMMA): L[n,c] = sum_f in[b,n,c,f]*outPrev[b,c,f] via delta-column B:
//   per c, B holds outPrev[c,f_k] only in column c%16; 16 c's x 4 K-chunks
//   chain into one accumulator per c-group. D layout: lane=j holds c=j (+16g),
//   vgpr r holds n-row r (low lanes) / r+8 (high lanes).
// PASS==1 stores L (priors) to Lbuf; PASS==2 preloads Lbuf as the C operand.
// Phase 2 (VALU): softmax over c (shfl_xor within 16-lane halves), then
//   Opart[c,f] += a[n,c]*in[n,c,f], lane j owning c=j and c=j+16.
// ---------------------------------------------------------------------------
template <int PASS>
__global__ __launch_bounds__(256) void k_route(const float* __restrict__ in,
                                               const float* __restrict__ outPrev,
                                               float* __restrict__ Lbuf,
                                               float* __restrict__ opart) {
  const int wid  = (blockIdx.x << 3) + (threadIdx.x >> 5);  // 0..2303
  const int lane = threadIdx.x & 31;
  const int j    = lane & 15;
  const bool low = lane < 16;
  const int b    = wid / SLOTS2;
  const int slot = wid % SLOTS2;
  const int fo   = low ? 0 : 2;  // K rows {0,1} on low lanes, {2,3} on high

  // Preload the delta-column B values: this lane is the active column for
  // c = j (group 0) and c = j+16 (group 1); it supplies f = 4q+fo, 4q+fo+1.
  v2f preB[2][4];
  {
    const float* op = outPrev + b * CF;
#pragma unroll
    for (int g = 0; g < 2; ++g) {
      const int c_act = j + 16 * g;
#pragma unroll
      for (int q = 0; q < 4; ++q)
        preB[g][q] = *(const v2f*)(op + c_act * FF + 4 * q + fo);
    }
  }

  v4f Oacc[2][4];
#pragma unroll
  for (int g = 0; g < 2; ++g)
#pragma unroll
    for (int q = 0; q < 4; ++q) Oacc[g][q] = (v4f){0.f, 0.f, 0.f, 0.f};

  for (int tt = 0; tt < 2; ++tt) {
    const int tile = slot * 2 + tt;
    const int n0 = tile * 16;

    v8f D[2];
    if (PASS == 2) {
      const float* lp = Lbuf + ((size_t)(b * TILES_PER_B + tile) * 2) * 256 + lane * 8;
      D[0] = *(const v8f*)(lp);
      D[1] = *(const v8f*)(lp + 256);
    } else {
      D[0] = (v8f){};
      D[1] = (v8f){};
    }

    // ---- Phase 1: WMMA dot products (A-slot of this lane is n-row j) ----
    const size_t rowA = (size_t)(b * NN + n0 + j) * CF;
#pragma unroll
    for (int g = 0; g < 2; ++g) {
      for (int cc = 0; cc < 16; ++cc) {
        const float* ap = in + rowA + (cc + 16 * g) * FF + fo;
#pragma unroll
        for (int q = 0; q < 4; ++q) {
          v2f Aop = *(const v2f*)(ap + 4 * q);
          v2f Bop = (j == cc) ? preB[g][q] : (v2f){0.f, 0.f};
          D[g] = WMMA_F32X4(Aop, Bop, D[g]);
        }
      }
    }

    if (PASS == 1) {
      float* lp = Lbuf + ((size_t)(b * TILES_PER_B + tile) * 2) * 256 + lane * 8;
      *(v8f*)(lp) = D[0];
      *(v8f*)(lp + 256) = D[1];
    }

    // ---- softmax over c per n-row (lanes within a 16-half share the row) ----
    float a0[8], a1[8];
#pragma unroll
    for (int r = 0; r < 8; ++r) {
      float m = fmaxf(D[0][r], D[1][r]);
#pragma unroll
      for (int mask = 1; mask < 16; mask <<= 1) m = fmaxf(m, __shfl_xor(m, mask));
      const float e0 = __expf(D[0][r] - m);
      const float e1 = __expf(D[1][r] - m);
      float s = e0 + e1;
#pragma unroll
      for (int mask = 1; mask < 16; mask <<= 1) s += __shfl_xor(s, mask);
      const float inv = 1.0f / s;
      a0[r] = e0 * inv;
      a1[r] = e1 * inv;
    }

    // ---- Phase 2: weighted accumulation (tile re-read is cache-hot) ----
#pragma unroll
    for (int r = 0; r < 8; ++r) {
      const int n = n0 + r + (low ? 0 : 8);
      const float* p0 = in + (size_t)(b * NN + n) * CF + j * FF;  // c = j
      const float* p1 = p0 + 16 * FF;                             // c = j+16
#pragma unroll
      for (int q = 0; q < 4; ++q) {
        Oacc[0][q] += a0[r] * (*(const v4f*)(p0 + 4 * q));
        Oacc[1][q] += a1[r] * (*(const v4f*)(p1 + 4 * q));
      }
    }
  }

  // combine the two n-halves (lanes j and j+16 own the same c)
#pragma unroll
  for (int g = 0; g < 2; ++g)
#pragma unroll
    for (int q = 0; q < 4; ++q) {
      v4f t = Oacc[g][q];
#pragma unroll
      for (int e = 0; e < 4; ++e) t[e] += __shfl_xor(t[e], 16);
      Oacc[g][q] = t;
    }

  if (low) {
#pragma unroll
    for (int g = 0; g < 2; ++g) {
      float* dst = opart + ((size_t)(b * SLOTS2 + slot) * CC + j + 16 * g) * FF;
#pragma unroll
      for (int q = 0; q < 4; ++q) *(v4f*)(dst + 4 * q) = Oacc[g][q];
    }
  }
}

__global__ __launch_bounds__(256) void k_reduce_out(const float* __restrict__ opart,
                                                    const float* __restrict__ bias,
                                                    float* __restrict__ out) {
  const int idx = blockIdx.x * 256 + threadIdx.x;  // 32768
  const int b = idx >> 9;
  const int cf = idx & 511;
  const float* p = opart + (size_t)b * SLOTS2 * CF + cf;
  float s = 0.0f;
  for (int k = 0; k < SLOTS2; ++k) s += p[k * CF];
  out[idx] = s + bias[cf];
}

// ---------------------------------------------------------------------------
// ws layout (floats): out0[32768] | out1[32768] | spart[262144]
//                     | Lbuf[2359296] | opart[1179648]  (~15.5 MB total)
// ---------------------------------------------------------------------------
extern "C" void kernel_launch(void* const* d_in, const int* in_sizes, int n_in,
                              void* d_out, int out_size, void* d_ws, size_t ws_size,
                              hipStream_t stream) {
  (void)in_sizes; (void)n_in; (void)out_size; (void)ws_size;
  const float* in   = (const float*)d_in[0];
  const float* bias = (const float*)d_in[1];
  float* out = (float*)d_out;

  float* ws    = (float*)d_ws;
  float* out0  = ws;
  float* out1  = ws + 32768;
  float* spart = ws + 65536;
  float* Lbuf  = ws + 65536 + 262144;
  float* opart = Lbuf + (size_t)BB * TILES_PER_B * 512;

  // Pass 1: uniform-softmax reduction (16384 waves -> HBM-latency friendly)
  k_sum<<<2048, 256, 0, stream>>>(in, spart);
  k_reduce_out0<<<128, 256, 0, stream>>>(spart, bias, out0);
  // Pass 2: iteration 1 (stores priors L1), inputs now L2-resident
  k_route<1><<<288, 256, 0, stream>>>(in, out0, Lbuf, opart);
  k_reduce_out<<<128, 256, 0, stream>>>(opart, bias, out1);
  // Pass 3: iteration 2 (L2 = L1 + dot), final output
  k_route<2><<<288, 256, 0, stream>>>(in, out1, Lbuf, opart);
  k_reduce_out<<<128, 256, 0, stream>>>(opart, bias, out);
}